// JointBilateralFilter_62801011802526
// MI455X (gfx1250) — compile-verified
//
#include <hip/hip_runtime.h>
#include <hip/hip_fp16.h>

// Joint bilateral filter, 5x5, B=1, C=3 guide (template), Cv=2 payload (vector).
// MI455X plan: bandwidth-bound stencil (~24 B/pixel -> ~2.1us at 23.3 TB/s).
// No matmul structure -> no WMMA; use CDNA5 async global->LDS staging, then
// strip-mined register reuse so each LDS texel is read once per thread.

#define IMG_W 1920
#define IMG_H 1080
#define IMG_HW (IMG_W * IMG_H)

#define TILE_W 64
#define TILE_H 16
#define HALO   2
#define GW (TILE_W + 2 * HALO)   // 68
#define GH (TILE_H + 2 * HALO)   // 20
#define NCELL (GW * GH)          // 1360
#define NTHREADS 256
#define ROWS_PER_THREAD 4        // output rows per thread
#define STRIP (ROWS_PER_THREAD + 4)  // 8 halo rows feeding those 4 outputs

#define KVAL   (-0.125f)   // -1/(2*sigma_spatial^2)
#define SIDIV  (50.0f)     //  1/(2*sigma_intensity^2)

// ---- CDNA5 async global -> LDS load of one dword (tracked by ASYNCcnt) ----
__device__ __forceinline__ void async_load_f32(const float* gsrc, float* ldst) {
#if defined(__HIP_DEVICE_COMPILE__)
#if __has_builtin(__builtin_amdgcn_global_load_async_to_lds_b32)
    __builtin_amdgcn_global_load_async_to_lds_b32(
        (__attribute__((address_space(1))) int*)(size_t)gsrc,
        (__attribute__((address_space(3))) int*)ldst,
        0, 0);
#else
    unsigned lofs = (unsigned)(size_t)(__attribute__((address_space(3))) int*)ldst;
    asm volatile("global_load_async_to_lds_b32 %0, %1, off"
                 :: "v"(lofs), "v"(gsrc)
                 : "memory");
#endif
#else
    (void)gsrc; (void)ldst;
#endif
}

__device__ __forceinline__ void wait_asynccnt0() {
#if defined(__HIP_DEVICE_COMPILE__)
#if __has_builtin(__builtin_amdgcn_s_wait_asynccnt)
    __builtin_amdgcn_s_wait_asynccnt(0);
#else
    asm volatile("s_wait_asynccnt 0" ::: "memory");
#endif
#endif
}

__global__ __launch_bounds__(NTHREADS)
void jbf_kernel(const float* __restrict__ tpl,   // (3,H,W) fp32
                const float* __restrict__ vec,   // (2,H,W) fp32
                __half* __restrict__ out)        // (2,H,W) fp16
{
    // Interleaved (T0,T1,T2,V0) per texel -> one ds_load_b128 per tap,
    // plus a separate V1 plane (ds_load_b32). 27,200 B of LDS per block.
    __shared__ __align__(16) float sPix4[NCELL * 4];
    __shared__ float sV1[NCELL];

    const int tx  = threadIdx.x;              // 0..63
    const int ty  = threadIdx.y;              // 0..3
    const int tid = ty * TILE_W + tx;
    const int x0  = blockIdx.x * TILE_W;
    const int y0  = blockIdx.y * TILE_H;

    // ---------------- stage haloed tile: 5 planes x GH x GW dwords -----------
    // In-bounds texels: async DMA global->LDS. Out-of-bounds: zero (SAME pad).
    for (int i = tid; i < 5 * NCELL; i += NTHREADS) {
        const int p  = i / NCELL;             // plane 0..4
        const int c  = i - p * NCELL;         // cell within tile
        const int ly = c / GW;
        const int lx = c - ly * GW;
        const int gx = x0 - HALO + lx;
        const int gy = y0 - HALO + ly;

        float* ldst = (p < 4) ? (sPix4 + c * 4 + p) : (sV1 + c);

        if ((unsigned)gx < (unsigned)IMG_W && (unsigned)gy < (unsigned)IMG_H) {
            const float* base = (p < 3) ? (tpl + p * IMG_HW)
                                        : (vec + (p - 3) * IMG_HW);
            async_load_f32(base + gy * IMG_W + gx, ldst);
        } else {
            *ldst = 0.0f;                     // plain ds_store, covered by barrier
        }
    }

    wait_asynccnt0();      // our async writes to LDS have landed
    __syncthreads();       // everyone's writes (ds + async) visible to all waves

    // ---------------- compute: strip-mined, one LDS read per texel -----------
    // Thread owns output rows yl0..yl0+3; their 5x5 windows span halo rows
    // yl0..yl0+7. Read each of the 8x5 window texels ONCE and scatter it into
    // every output row whose window contains it (dy = j - rr in [0,4]).
    const float4* pix = (const float4*)sPix4;
    const int yl0 = ty * ROWS_PER_THREAD;

    float4 cp[ROWS_PER_THREAD];
#pragma unroll
    for (int rr = 0; rr < ROWS_PER_THREAD; ++rr)
        cp[rr] = pix[(yl0 + rr + HALO) * GW + (tx + HALO)];

    float n0[ROWS_PER_THREAD], n1[ROWS_PER_THREAD], den[ROWS_PER_THREAD];
#pragma unroll
    for (int rr = 0; rr < ROWS_PER_THREAD; ++rr) { n0[rr] = n1[rr] = den[rr] = 0.0f; }

#pragma unroll
    for (int j = 0; j < STRIP; ++j) {         // halo row yl0 + j
        const int rowoff = (yl0 + j) * GW + tx;
#pragma unroll
        for (int dx = 0; dx < 5; ++dx) {
            const float4 q  = pix[rowoff + dx];   // T0,T1,T2,V0
            const float  v1 = sV1[rowoff + dx];
#pragma unroll
            for (int rr = 0; rr < ROWS_PER_THREAD; ++rr) {
                const int dy = j - rr;            // compile-time after unroll
                if (dy >= 0 && dy <= 4) {
                    const float d0 = cp[rr].x - q.x;
                    const float d1 = cp[rr].y - q.y;
                    const float d2 = cp[rr].z - q.z;
                    const float id = fmaf(d2, d2, fmaf(d1, d1, d0 * d0));
                    // clip(1 - |KVAL - id*SIDIV|, 0, 1); id >= 0 so this is
                    // sat(1 - (0.125 + 50*id)) -> one v_sub with clamp.
                    const float wgt =
                        __saturatef(1.0f - fabsf(KVAL - id * SIDIV));
                    n0[rr]  = fmaf(q.w, wgt, n0[rr]);   // V0 accumulation
                    n1[rr]  = fmaf(v1,  wgt, n1[rr]);   // V1 accumulation
                    den[rr] += wgt;
                }
            }
        }
    }

    const int x = x0 + tx;                    // always < IMG_W (1920 % 64 == 0)
#pragma unroll
    for (int rr = 0; rr < ROWS_PER_THREAD; ++rr) {
        const int y = y0 + yl0 + rr;
        if (y < IMG_H) {
            const float r = __builtin_amdgcn_rcpf(den[rr]);  // den > 0 always
            out[0 * IMG_HW + y * IMG_W + x] = __float2half(n0[rr] * r);
            out[1 * IMG_HW + y * IMG_W + x] = __float2half(n1[rr] * r);
        }
    }
}

extern "C" void kernel_launch(void* const* d_in, const int* in_sizes, int n_in,
                              void* d_out, int out_size, void* d_ws, size_t ws_size,
                              hipStream_t stream) {
    (void)in_sizes; (void)n_in; (void)out_size; (void)d_ws; (void)ws_size;
    const float* tpl = (const float*)d_in[0];   // (1,3,1080,1920) fp32
    const float* vec = (const float*)d_in[1];   // (1,2,1080,1920) fp32
    __half* out = (__half*)d_out;               // (1,2,1080,1920) fp16

    dim3 block(TILE_W, NTHREADS / TILE_W);                    // 64 x 4 = 8 waves
    dim3 grid(IMG_W / TILE_W, (IMG_H + TILE_H - 1) / TILE_H); // 30 x 68
    jbf_kernel<<<grid, block, 0, stream>>>(tpl, vec, out);
}